// DynamicVFE_31550829757175
// MI455X (gfx1250) — compile-verified
//
#include <hip/hip_runtime.h>

// ---------------- problem constants (from reference) ----------------
#define NXG 352
#define NYG 400
#define NZG 1
#define NBATCH 2
#define NCANVAS (NBATCH * NZG * NYG * NXG)   // 281600
#define NPTS 400000
#define BN_EPS 1e-3f

typedef __attribute__((ext_vector_type(2))) float v2f;
typedef __attribute__((ext_vector_type(8))) float v8f;

// -------- workspace layout (float elements); total ~101.44M floats (~406MB) ----
constexpr size_t OFF_VSUM = 0;                                // NCANVAS*4
constexpr size_t OFF_VCNT = OFF_VSUM + (size_t)NCANVAS * 4;   // NCANVAS
constexpr size_t OFF_IDX  = OFF_VCNT + (size_t)NCANVAS;       // NPTS ints
constexpr size_t OFF_FEAT = OFF_IDX + (size_t)NPTS;           // NPTS*12 (K padded 10->12)
constexpr size_t OFF_Y0   = OFF_FEAT + (size_t)NPTS * 12;     // NPTS*64
constexpr size_t OFF_V0   = OFF_Y0 + (size_t)NPTS * 64;       // NCANVAS*64
constexpr size_t OFF_Y1   = OFF_V0 + (size_t)NCANVAS * 64;    // NPTS*128
constexpr size_t OFF_BN   = OFF_Y1 + (size_t)NPTS * 128;      // 768 floats of stats/params

__device__ __forceinline__ void atomic_fmax_nonneg(float* p, float v) {
    // valid because all stored values are >= 0 (post-ReLU) and canvas is 0-initialized
    atomicMax((unsigned int*)p, __float_as_uint(v));
}

// ---------- pass A: voxel index + scatter-mean accumulation ----------
__global__ void k_mean_acc(const float* __restrict__ feats, const int* __restrict__ coors,
                           float* __restrict__ vsum, float* __restrict__ vcnt,
                           int* __restrict__ idxbuf, int n) {
    int i = blockIdx.x * blockDim.x + threadIdx.x;
    if (i >= n) return;
    int b = coors[i * 4 + 0], z = coors[i * 4 + 1];
    int y = coors[i * 4 + 2], x = coors[i * 4 + 3];
    int id = ((b * NZG + z) * NYG + y) * NXG + x;
    idxbuf[i] = id;
    atomicAdd(&vsum[id * 4 + 0], feats[i * 4 + 0]);
    atomicAdd(&vsum[id * 4 + 1], feats[i * 4 + 1]);
    atomicAdd(&vsum[id * 4 + 2], feats[i * 4 + 2]);
    atomicAdd(&vsum[id * 4 + 3], feats[i * 4 + 3]);
    atomicAdd(&vcnt[id], 1.0f);
}

// ---------- pass B: build padded 12-wide point features (float4 stores) ----------
__global__ void k_build_feat(const float* __restrict__ feats, const int* __restrict__ coors,
                             const float* __restrict__ vsum, const float* __restrict__ vcnt,
                             const int* __restrict__ idxbuf, float* __restrict__ feat, int n) {
    int i = blockIdx.x * blockDim.x + threadIdx.x;
    if (i >= n) return;
    int id = idxbuf[i];
    float f0 = feats[i * 4 + 0], f1 = feats[i * 4 + 1];
    float f2 = feats[i * 4 + 2], f3 = feats[i * 4 + 3];
    float inv = 1.0f / fmaxf(vcnt[id], 1.0f);
    float m0 = vsum[id * 4 + 0] * inv;
    float m1 = vsum[id * 4 + 1] * inv;
    float m2 = vsum[id * 4 + 2] * inv;
    float cz = (float)coors[i * 4 + 1];
    float cy = (float)coors[i * 4 + 2];
    float cx = (float)coors[i * 4 + 3];
    float4* o = (float4*)(feat + (size_t)i * 12);   // 48B row stride keeps 16B alignment
    o[0] = make_float4(f0, f1, f2, f3);
    o[1] = make_float4(f0 - m0, f1 - m1, f2 - m2,
                       f0 - (cx * 0.2f + (0.1f + 0.0f)));
    o[2] = make_float4(f1 - (cy * 0.2f + (0.1f - 40.0f)),
                       f2 - (cz * 4.0f + (2.0f - 3.0f)), 0.0f, 0.0f);
}

// ---------- GEMM0: (N,12pad) @ W0(10->12pad x 64) via v_wmma_f32_16x16x4_f32 ----------
__global__ __launch_bounds__(256) void k_gemm0(const float* __restrict__ feat,
                                               const float* __restrict__ W0,
                                               float* __restrict__ y0, int n) {
    __shared__ float ldsW[12 * 64];  // 6 pairs * 64 cols * 2
    for (int i = threadIdx.x; i < 768; i += 256) {
        int sub = i & 1, col = (i >> 1) & 63, pair = i >> 7;
        int k = pair * 2 + sub;
        ldsW[i] = (k < 10) ? W0[k * 64 + col] : 0.0f;
    }
    __syncthreads();

    int wave = threadIdx.x >> 5, lane = threadIdx.x & 31;
    int row0 = blockIdx.x * 128 + wave * 16;        // wave-uniform: EXEC all-1s in wave
    if (row0 >= n) return;
    int half = lane >> 4, lcol = lane & 15;
    int mrow = row0 + lcol; if (mrow >= n) mrow = n - 1;

    v8f acc[4];
#pragma unroll
    for (int t = 0; t < 4; ++t) acc[t] = (v8f)(0.0f);

#pragma unroll
    for (int kp = 0; kp < 3; ++kp) {  // K = 12 in steps of 4
        v2f a = *(const v2f*)(feat + (size_t)mrow * 12 + kp * 4 + half * 2);
#pragma unroll
        for (int t = 0; t < 4; ++t) {
            v2f b = *(const v2f*)(&ldsW[((kp * 2 + half) * 64 + (t * 16 + lcol)) * 2]);
            acc[t] = __builtin_amdgcn_wmma_f32_16x16x4_f32(
                false, a, false, b, (short)0, acc[t], false, false);
        }
    }
    // D layout: VGPR r holds M = r + 8*half, N = lcol (+16*t)
    if (row0 + 16 <= n) {            // wave-uniform fast path: no per-store exec churn
#pragma unroll
        for (int t = 0; t < 4; ++t)
#pragma unroll
            for (int r = 0; r < 8; ++r)
                y0[(size_t)(row0 + half * 8 + r) * 64 + t * 16 + lcol] = acc[t][r];
    } else {                         // never taken for N multiple of 16; kept for safety
#pragma unroll
        for (int t = 0; t < 4; ++t)
#pragma unroll
            for (int r = 0; r < 8; ++r) {
                int row = row0 + half * 8 + r;
                if (row < n) y0[(size_t)row * 64 + t * 16 + lcol] = acc[t][r];
            }
    }
}

// ---------- per-channel sum / sumsq over N rows (for batch-stat BN) ----------
__global__ void k_colstats(const float* __restrict__ y, int n, int C,
                           float* __restrict__ sum, float* __restrict__ sq) {
    __shared__ float ssum[256];
    __shared__ float ssq[256];
    int col = threadIdx.x % C;
    int rgroup = threadIdx.x / C;
    int groups = 256 / C;
    float s = 0.0f, q = 0.0f;
    for (int r = blockIdx.x * groups + rgroup; r < n; r += gridDim.x * groups) {
        float v = y[(size_t)r * C + col];
        s += v; q += v * v;
    }
    ssum[threadIdx.x] = s; ssq[threadIdx.x] = q;
    __syncthreads();
    if (threadIdx.x < C) {
        for (int g = 1; g < groups; ++g) { s += ssum[g * C + col]; q += ssq[g * C + col]; }
        atomicAdd(&sum[col], s);
        atomicAdd(&sq[col], q);
    }
}

__global__ void k_bnparams(const float* __restrict__ sum, const float* __restrict__ sq,
                           const float* __restrict__ g, const float* __restrict__ b,
                           float* __restrict__ scale, float* __restrict__ shift,
                           int C, float invn) {
    int c = blockIdx.x * blockDim.x + threadIdx.x;
    if (c >= C) return;
    float mu = sum[c] * invn;
    float var = sq[c] * invn - mu * mu;          // biased var, matches jnp.var
    float sc = g[c] * rsqrtf(var + BN_EPS);
    scale[c] = sc;
    shift[c] = b[c] - mu * sc;
}

// ---------- BN + ReLU apply, optional store, atomic scatter-max (4 ch / thread) ----------
__global__ void k_bnmax4(const float4* __restrict__ y, const float* __restrict__ scale,
                         const float* __restrict__ shift, const int* __restrict__ idxbuf,
                         float4* __restrict__ pf, float* __restrict__ vmax,
                         int total4, int log2C) {
    int e4 = blockIdx.x * blockDim.x + threadIdx.x;
    if (e4 >= total4) return;
    int C = 1 << log2C;
    int base = e4 * 4;
    int c = base & (C - 1);          // 4 consecutive channels share a row
    int r = base >> log2C;
    float4 v = y[e4];
    v.x = fmaxf(v.x * scale[c + 0] + shift[c + 0], 0.0f);
    v.y = fmaxf(v.y * scale[c + 1] + shift[c + 1], 0.0f);
    v.z = fmaxf(v.z * scale[c + 2] + shift[c + 2], 0.0f);
    v.w = fmaxf(v.w * scale[c + 3] + shift[c + 3], 0.0f);
    if (pf) pf[e4] = v;
    float* dst = &vmax[(size_t)idxbuf[r] * C + c];
    atomic_fmax_nonneg(dst + 0, v.x);
    atomic_fmax_nonneg(dst + 1, v.y);
    atomic_fmax_nonneg(dst + 2, v.z);
    atomic_fmax_nonneg(dst + 3, v.w);
}

// ---------- GEMM1: concat(pf0, v0[idx]) (N,128) @ W1(128x128) ----------
__global__ __launch_bounds__(256) void k_gemm1(const float* __restrict__ pf0,
                                               const float* __restrict__ v0,
                                               const int* __restrict__ idxbuf,
                                               const float* __restrict__ W1,
                                               float* __restrict__ y1, int n) {
    __shared__ float ldsW[128 * 128];  // 64 KB, K-pairs packed: [(pair*128+col)*2 + (k&1)]
    for (int i = threadIdx.x; i < 16384; i += 256) {
        int sub = i & 1, col = (i >> 1) & 127, pair = i >> 8;
        ldsW[i] = W1[(pair * 2 + sub) * 128 + col];
    }
    __syncthreads();

    int wave = threadIdx.x >> 5, lane = threadIdx.x & 31;
    int row0 = blockIdx.x * 128 + wave * 16;        // wave-uniform guard
    if (row0 >= n) return;
    int half = lane >> 4, lcol = lane & 15;
    int mrow = row0 + lcol; if (mrow >= n) mrow = n - 1;
    int vrow = idxbuf[mrow];
    const float* arow0 = pf0 + (size_t)mrow * 64;          // channels   0..63
    const float* arow1 = v0  + (size_t)vrow * 64;          // channels  64..127

    v8f acc[8];
#pragma unroll
    for (int t = 0; t < 8; ++t) acc[t] = (v8f)(0.0f);

    for (int kp = 0; kp < 32; ++kp) {  // K = 128 in steps of 4
        int kA = kp * 4 + half * 2;
        v2f a = (kA < 64) ? *(const v2f*)(arow0 + kA)
                          : *(const v2f*)(arow1 + (kA - 64));
#pragma unroll
        for (int t = 0; t < 8; ++t) {
            v2f b = *(const v2f*)(&ldsW[((kp * 2 + half) * 128 + (t * 16 + lcol)) * 2]);
            acc[t] = __builtin_amdgcn_wmma_f32_16x16x4_f32(
                false, a, false, b, (short)0, acc[t], false, false);
        }
    }
    if (row0 + 16 <= n) {            // wave-uniform fast path
#pragma unroll
        for (int t = 0; t < 8; ++t)
#pragma unroll
            for (int r = 0; r < 8; ++r)
                y1[(size_t)(row0 + half * 8 + r) * 128 + t * 16 + lcol] = acc[t][r];
    } else {
#pragma unroll
        for (int t = 0; t < 8; ++t)
#pragma unroll
            for (int r = 0; r < 8; ++r) {
                int row = row0 + half * 8 + r;
                if (row < n) y1[(size_t)row * 128 + t * 16 + lcol] = acc[t][r];
            }
    }
}

// =============================== launcher ===============================
extern "C" void kernel_launch(void* const* d_in, const int* in_sizes, int n_in,
                              void* d_out, int out_size, void* d_ws, size_t ws_size,
                              hipStream_t stream) {
    const float* features = (const float*)d_in[0];
    const int*   coors    = (const int*)d_in[1];
    const float* W0       = (const float*)d_in[2];
    const float* g0       = (const float*)d_in[3];
    const float* b0       = (const float*)d_in[4];
    const float* W1       = (const float*)d_in[5];
    const float* g1       = (const float*)d_in[6];
    const float* b1       = (const float*)d_in[7];
    float* out = (float*)d_out;
    float* ws  = (float*)d_ws;

    int n = in_sizes[0] / 4;

    float* vsum = ws + OFF_VSUM;
    float* vcnt = ws + OFF_VCNT;
    int*   idxb = (int*)(ws + OFF_IDX);
    float* feat = ws + OFF_FEAT;
    float* y0   = ws + OFF_Y0;
    float* v0   = ws + OFF_V0;
    float* y1   = ws + OFF_Y1;
    float* bn   = ws + OFF_BN;
    float* bn0sum = bn;         float* bn0sq = bn + 64;
    float* bn1sum = bn + 128;   float* bn1sq = bn + 256;
    float* bn0scale = bn + 384; float* bn0shift = bn + 448;
    float* bn1scale = bn + 512; float* bn1shift = bn + 640;

    // zero atomic-accumulation targets (graph-capture-safe memsets)
    hipMemsetAsync(vsum, 0, (size_t)NCANVAS * 5 * sizeof(float), stream);  // vsum+vcnt
    hipMemsetAsync(v0,   0, (size_t)NCANVAS * 64 * sizeof(float), stream);
    hipMemsetAsync(bn,   0, 384 * sizeof(float), stream);
    hipMemsetAsync(out,  0, (size_t)out_size * sizeof(float), stream);

    int nb = (n + 255) / 256;
    k_mean_acc<<<nb, 256, 0, stream>>>(features, coors, vsum, vcnt, idxb, n);
    k_build_feat<<<nb, 256, 0, stream>>>(features, coors, vsum, vcnt, idxb, feat, n);

    int gb = (n + 127) / 128;
    k_gemm0<<<gb, 256, 0, stream>>>(feat, W0, y0, n);

    k_colstats<<<512, 256, 0, stream>>>(y0, n, 64, bn0sum, bn0sq);
    k_bnparams<<<1, 128, 0, stream>>>(bn0sum, bn0sq, g0, b0, bn0scale, bn0shift, 64, 1.0f / n);
    k_bnmax4<<<(n * 16 + 255) / 256, 256, 0, stream>>>((const float4*)y0, bn0scale, bn0shift,
                                                       idxb, (float4*)y0 /*pf0 in place*/,
                                                       v0, n * 16, 6);

    k_gemm1<<<gb, 256, 0, stream>>>(y0, v0, idxb, W1, y1, n);

    k_colstats<<<512, 256, 0, stream>>>(y1, n, 128, bn1sum, bn1sq);
    k_bnparams<<<1, 128, 0, stream>>>(bn1sum, bn1sq, g1, b1, bn1scale, bn1shift, 128, 1.0f / n);
    k_bnmax4<<<(n * 32 + 255) / 256, 256, 0, stream>>>((const float4*)y1, bn1scale, bn1shift,
                                                       idxb, nullptr, out, n * 32, 7);

    (void)n_in; (void)ws_size;
}